// Classifier_16716012716288
// MI455X (gfx1250) — compile-verified
//
#include <hip/hip_runtime.h>
#include <hip/hip_bf16.h>
#include <math.h>
#include <stdint.h>

// Problem constants from the reference
#define B_   16
#define T_   512
#define E_   256
#define H_   256
#define HEADS_ 8
#define NL_  3
#define NC_  10

typedef float v2f __attribute__((ext_vector_type(2)));
typedef float v8f __attribute__((ext_vector_type(8)));

// ---------------------------------------------------------------------------
// 1) Embedding gather: cat[:, 0:256] = emb[x]
// ---------------------------------------------------------------------------
__global__ void embed_kernel(const int* __restrict__ x,
                             const float* __restrict__ emb,
                             float* __restrict__ cat) {
    int row = blockIdx.x * 4 + (threadIdx.x >> 6);   // 4 rows per 256-thread block
    int l   = threadIdx.x & 63;                       // 64 lanes * float4 = 256 floats
    int tok = x[row];
    const float4* src = (const float4*)(emb + (long)tok * E_);
    float4*       dst = (float4*)(cat + (long)row * (2 * E_));
    dst[l] = src[l];
}

// ---------------------------------------------------------------------------
// 2) Layer mix (fused, streaming, float4): M[b,j,i] = sum_l w*G + (1-w)*Gp
// ---------------------------------------------------------------------------
__global__ void mix_kernel(const float* __restrict__ G,
                           const float* __restrict__ Gp,
                           const float* __restrict__ mix_w,
                           float* __restrict__ Mbuf) {
    const long TT4 = (long)T_ * T_ / 4;
    long i4 = (long)blockIdx.x * blockDim.x + threadIdx.x;
    if (i4 >= (long)B_ * TT4) return;
    long b = i4 / TT4, rem = i4 - b * TT4;
    const float4* G4  = (const float4*)G;
    const float4* Gp4 = (const float4*)Gp;
    float4 acc = make_float4(0.f, 0.f, 0.f, 0.f);
#pragma unroll
    for (int l = 0; l < NL_; ++l) {
        float w = mix_w[l];
        float4 g  = G4 [(b * NL_ + l) * TT4 + rem];
        float4 gp = Gp4[(b * NL_ + l) * TT4 + rem];
        acc.x += w * g.x + (1.f - w) * gp.x;
        acc.y += w * g.y + (1.f - w) * gp.y;
        acc.z += w * g.z + (1.f - w) * gp.z;
        acc.w += w * g.w + (1.f - w) * gp.w;
    }
    ((float4*)Mbuf)[i4] = acc;
}

// ---------------------------------------------------------------------------
// Generic f32 WMMA GEMM with async-to-LDS staged B operand.
//   C = A @ B (row-major). ATRANS: A element (m,k) stored at A[k*lda + m].
// Block: 256 threads = 8 waves. Block tile 64x128; each wave owns a 16x64
// strip (4 accumulators, A-fragment reused 4x). B tile (32x128 per K-chunk)
// is staged through LDS with global_load_async_to_lds_b128, double buffered
// and pipelined with s_wait_asynccnt.
// grid = (M/64, N/128, batch); M%64==0, N%128==0, K%32==0 -> full EXEC.
// ---------------------------------------------------------------------------
template <bool ATRANS>
__global__ __launch_bounds__(256) void wmma_gemm_kernel(
    const float* __restrict__ A, const float* __restrict__ Bm,
    float* __restrict__ C, int K, int lda, int ldb, int ldc,
    long strideA, long strideB, long strideC) {
    __shared__ __align__(16) float sB[2 * 32 * 128];   // 32 KB double buffer
    long batch = blockIdx.z;
    A  += batch * strideA;
    Bm += batch * strideB;
    C  += batch * strideC;

    int tid  = threadIdx.x;
    int wave = tid >> 5;
    int lane = tid & 31;
    int half = lane >> 4;
    int idx  = lane & 15;
    int wr   = wave >> 1;            // 0..3 row-group
    int wc   = wave & 1;             // 0..1 col-group
    int row0    = blockIdx.x * 64 + wr * 16;
    int colBase = blockIdx.y * 128;
    int colLoc  = wc * 64 + idx;     // column within 128-wide LDS tile

    // Async staging layout: thread t copies 4x float4 of row (t>>3)
    int crow = tid >> 3;             // 0..31 (chunk-local k row)
    int cq   = tid & 7;              // quad slot within row
    const float* gB0 = Bm + (long)crow * ldb + colBase + cq * 4;
    unsigned ldsRow0 = (unsigned)(uintptr_t)(&sB[0]) +
                       (unsigned)(crow * 128 + cq * 4) * 4u;

    int nchunks = K >> 5;

    // Prologue: chunk 0 -> buffer 0
#pragma unroll
    for (int j = 0; j < 4; ++j)
        asm volatile("global_load_async_to_lds_b128 %0, %1, off"
                     :: "v"(ldsRow0 + (unsigned)(j * 128)),
                        "v"((unsigned long long)(uintptr_t)(gB0 + j * 32))
                     : "memory");

    v8f acc0 = {}, acc1 = {}, acc2 = {}, acc3 = {};

    for (int c = 0; c < nchunks; ++c) {
        if (c + 1 < nchunks) {
            const float* g = gB0 + (long)((c + 1) << 5) * ldb;
            unsigned l = ldsRow0 + (((c + 1) & 1) ? 16384u : 0u);
#pragma unroll
            for (int j = 0; j < 4; ++j)
                asm volatile("global_load_async_to_lds_b128 %0, %1, off"
                             :: "v"(l + (unsigned)(j * 128)),
                                "v"((unsigned long long)(uintptr_t)(g + j * 32))
                             : "memory");
            // chunk c complete (4 newer ops may remain in flight)
            asm volatile("s_wait_asynccnt 0x4" ::: "memory");
        } else {
            asm volatile("s_wait_asynccnt 0x0" ::: "memory");
        }
        __syncthreads();

        const float* bbuf = &sB[(c & 1) ? 4096 : 0];
        int kk = c << 5;
#pragma unroll
        for (int k8 = 0; k8 < 8; ++k8) {
            int kl = k8 * 4 + half * 2;      // chunk-local k
            int k0 = kk + kl;                // global k
            v2f a;
            if (ATRANS) {
                a.x = A[(long)(k0    ) * lda + row0 + idx];
                a.y = A[(long)(k0 + 1) * lda + row0 + idx];
            } else {
                const float* ap = A + (long)(row0 + idx) * lda + k0;
                a.x = ap[0];
                a.y = ap[1];
            }
            v2f b0, b1, b2, b3;
            b0.x = bbuf[(kl    ) * 128 + colLoc];
            b0.y = bbuf[(kl + 1) * 128 + colLoc];
            b1.x = bbuf[(kl    ) * 128 + colLoc + 16];
            b1.y = bbuf[(kl + 1) * 128 + colLoc + 16];
            b2.x = bbuf[(kl    ) * 128 + colLoc + 32];
            b2.y = bbuf[(kl + 1) * 128 + colLoc + 32];
            b3.x = bbuf[(kl    ) * 128 + colLoc + 48];
            b3.y = bbuf[(kl + 1) * 128 + colLoc + 48];
            acc0 = __builtin_amdgcn_wmma_f32_16x16x4_f32(false, a, false, b0, (short)0, acc0, false, false);
            acc1 = __builtin_amdgcn_wmma_f32_16x16x4_f32(false, a, false, b1, (short)0, acc1, false, false);
            acc2 = __builtin_amdgcn_wmma_f32_16x16x4_f32(false, a, false, b2, (short)0, acc2, false, false);
            acc3 = __builtin_amdgcn_wmma_f32_16x16x4_f32(false, a, false, b3, (short)0, acc3, false, false);
        }
        __syncthreads();   // all waves done reading before this buffer is rewritten
    }

    int colOut = colBase + wc * 64 + idx;
#pragma unroll
    for (int v = 0; v < 8; ++v) {
        long r = (long)(row0 + v + half * 8) * ldc;
        C[r + colOut]      = acc0[v];
        C[r + colOut + 16] = acc1[v];
        C[r + colOut + 32] = acc2[v];
        C[r + colOut + 48] = acc3[v];
    }
}

// ---------------------------------------------------------------------------
// Weight transpose (small matrices, done once per call)
// ---------------------------------------------------------------------------
__global__ void transpose_kernel(const float* __restrict__ in,
                                 float* __restrict__ out, int rows, int cols) {
    int i = blockIdx.x * blockDim.x + threadIdx.x;
    if (i >= rows * cols) return;
    int r = i / cols, c = i - r * cols;
    out[c * rows + r] = in[i];
}

// ---------------------------------------------------------------------------
// SwiGLU gating: next_in = t1 * sigmoid(t2)
// ---------------------------------------------------------------------------
__global__ void swiglu_kernel(const float* __restrict__ t1,
                              const float* __restrict__ t2,
                              float* __restrict__ out, int n) {
    int i = blockIdx.x * blockDim.x + threadIdx.x;
    if (i < n) out[i] = t1[i] * (1.f / (1.f + expf(-t2[i])));
}

// ---------------------------------------------------------------------------
// GRU scan: one persistent workgroup (24 waves), h + gate pre-acts in LDS.
// Per step: hg(16x768) = h(16x256) @ Whht(256x768) via f32 WMMA, then gates.
// ---------------------------------------------------------------------------
__global__ __launch_bounds__(768) void gru_scan_kernel(
    const float* __restrict__ xg,   // (B,T,3H)
    const float* __restrict__ Whht, // (H, 3H) = W_hh^T
    const float* __restrict__ b_ih, const float* __restrict__ b_hh,
    const float* __restrict__ h0,
    float* __restrict__ outputs,    // (B,T,H)
    float* __restrict__ hn) {       // (B,H)
    __shared__ float h_s [B_ * H_];       // 16 KB
    __shared__ float hg_s[B_ * 3 * H_];   // 48 KB
    int tid = threadIdx.x;
    for (int e = tid; e < B_ * H_; e += 768) h_s[e] = h0[e];
    __syncthreads();

    int wave = tid >> 5, lane = tid & 31, half = lane >> 4, idx = lane & 15;

    for (int t = 0; t < T_; ++t) {
        // hg = h @ Whht : 48 N-tiles over 24 waves, 2 tiles each
#pragma unroll
        for (int s = 0; s < 2; ++s) {
            int col0 = wave * 32 + s * 16;
            v8f acc = {};
            for (int kk = 0; kk < H_; kk += 4) {
                int k0 = kk + half * 2;
                v2f a;
                a.x = h_s[idx * H_ + k0];
                a.y = h_s[idx * H_ + k0 + 1];
                v2f b;
                b.x = Whht[(k0    ) * (3 * H_) + col0 + idx];
                b.y = Whht[(k0 + 1) * (3 * H_) + col0 + idx];
                acc = __builtin_amdgcn_wmma_f32_16x16x4_f32(false, a, false, b,
                                                            (short)0, acc, false, false);
            }
#pragma unroll
            for (int v = 0; v < 8; ++v)
                hg_s[(v + half * 8) * (3 * H_) + col0 + idx] = acc[v];
        }
        __syncthreads();

        // elementwise gates: 4096 elements over 768 threads
        for (int e = tid; e < B_ * H_; e += 768) {
            int bb = e >> 8, hi = e & 255;
            const float* xrow = xg + (long)(bb * T_ + t) * (3 * H_);
            float xr = xrow[hi]            + b_ih[hi];
            float xz = xrow[hi + H_]       + b_ih[hi + H_];
            float xn = xrow[hi + 2 * H_]   + b_ih[hi + 2 * H_];
            float hr = hg_s[bb * 3 * H_ + hi]           + b_hh[hi];
            float hz = hg_s[bb * 3 * H_ + hi + H_]      + b_hh[hi + H_];
            float hv = hg_s[bb * 3 * H_ + hi + 2 * H_]  + b_hh[hi + 2 * H_];
            float r = 1.f / (1.f + expf(-(xr + hr)));
            float z = 1.f / (1.f + expf(-(xz + hz)));
            float n = tanhf(xn + r * hv);
            float hnew = (1.f - z) * n + z * h_s[e];
            h_s[e] = hnew;
            outputs[(long)(bb * T_ + t) * H_ + hi] = hnew;
        }
        __syncthreads();
    }
    for (int e = tid; e < B_ * H_; e += 768) hn[e] = h_s[e];
}

// ---------------------------------------------------------------------------
// tanh(x + b1) elementwise (col bias, ld = 256)
// ---------------------------------------------------------------------------
__global__ void tanh_bias_kernel(float* __restrict__ t,
                                 const float* __restrict__ b1, int n) {
    int i = blockIdx.x * blockDim.x + threadIdx.x;
    if (i < n) t[i] = tanhf(t[i] + b1[i & (H_ - 1)]);
}

// ---------------------------------------------------------------------------
// Head projection: abuf[b,t,head] = dot(t3[b,t,:], W2[head,:]) + b2[head]
// ---------------------------------------------------------------------------
__global__ void head_proj_kernel(const float* __restrict__ t3,
                                 const float* __restrict__ W2,
                                 const float* __restrict__ b2,
                                 float* __restrict__ abuf) {
    int gid = blockIdx.x * blockDim.x + threadIdx.x; // B*T*HEADS threads
    int row = gid >> 3, head = gid & 7;
    const float* tr = t3 + (long)row * H_;
    const float* w  = W2 + head * H_;
    float acc = 0.f;
    for (int k = 0; k < H_; ++k) acc += tr[k] * w[k];
    abuf[gid] = acc + b2[head];
}

// ---------------------------------------------------------------------------
// Masked softmax over t, writing attention transposed to (B,HEADS,T)
// ---------------------------------------------------------------------------
__global__ __launch_bounds__(256) void softmax_kernel(
    const float* __restrict__ abuf, const int* __restrict__ lengths,
    float* __restrict__ attention) {
    int b = blockIdx.x >> 3, head = blockIdx.x & 7;
    int tid = threadIdx.x;
    int len = lengths[b];
    __shared__ float red[256];
    float v0 = (tid       < len) ? abuf[(b * T_ + tid      ) * HEADS_ + head] : -1e30f;
    float v1 = (tid + 256 < len) ? abuf[(b * T_ + tid + 256) * HEADS_ + head] : -1e30f;
    red[tid] = fmaxf(v0, v1);
    __syncthreads();
    for (int s = 128; s > 0; s >>= 1) {
        if (tid < s) red[tid] = fmaxf(red[tid], red[tid + s]);
        __syncthreads();
    }
    float mx = red[0];
    __syncthreads();
    float e0 = expf(v0 - mx), e1 = expf(v1 - mx);
    red[tid] = e0 + e1;
    __syncthreads();
    for (int s = 128; s > 0; s >>= 1) {
        if (tid < s) red[tid] += red[tid + s];
        __syncthreads();
    }
    float inv = 1.f / red[0];
    attention[(b * HEADS_ + head) * T_ + tid      ] = e0 * inv;
    attention[(b * HEADS_ + head) * T_ + tid + 256] = e1 * inv;
}

// ---------------------------------------------------------------------------
// Pool + classifier: avg[b,h] = sum_{t} (sum_head attn[b,head,t]) * out[b,t,h]/8
// then output[b,c] = avg . Wf[c] + bf[c]
// ---------------------------------------------------------------------------
__global__ __launch_bounds__(256) void finalize_kernel(
    const float* __restrict__ attention, const float* __restrict__ outputs,
    const float* __restrict__ Wf, const float* __restrict__ bf,
    float* __restrict__ outFinal) {
    int b = blockIdx.x;
    int tid = threadIdx.x;
    __shared__ float wsum[T_];
    __shared__ float avg_s[H_];
    for (int t = tid; t < T_; t += 256) {
        float s = 0.f;
#pragma unroll
        for (int head = 0; head < HEADS_; ++head)
            s += attention[(b * HEADS_ + head) * T_ + t];
        wsum[t] = s;
    }
    __syncthreads();
    float acc = 0.f;
    for (int t = 0; t < T_; ++t)
        acc += wsum[t] * outputs[(long)(b * T_ + t) * H_ + tid];
    avg_s[tid] = acc * (1.f / HEADS_);
    __syncthreads();
    if (tid < NC_) {
        float o = 0.f;
        for (int h = 0; h < H_; ++h) o += avg_s[h] * Wf[tid * H_ + h];
        outFinal[b * NC_ + tid] = o + bf[tid];
    }
}

// ---------------------------------------------------------------------------
extern "C" void kernel_launch(void* const* d_in, const int* in_sizes, int n_in,
                              void* d_out, int out_size, void* d_ws, size_t ws_size,
                              hipStream_t stream) {
    (void)in_sizes; (void)n_in; (void)out_size; (void)ws_size;
    const int*   x       = (const int*)  d_in[0];
    const int*   lengths = (const int*)  d_in[1];
    const float* h0      = (const float*)d_in[2];
    const float* emb     = (const float*)d_in[3];
    const float* G       = (const float*)d_in[4];
    const float* Gp      = (const float*)d_in[5];
    const float* mix_w   = (const float*)d_in[6];
    const float* Wc1     = (const float*)d_in[7];
    const float* Wc2     = (const float*)d_in[8];
    const float* W_ih    = (const float*)d_in[9];
    const float* W_hh    = (const float*)d_in[10];
    const float* b_ih    = (const float*)d_in[11];
    const float* b_hh    = (const float*)d_in[12];
    const float* W1      = (const float*)d_in[13];
    const float* b1      = (const float*)d_in[14];
    const float* W2      = (const float*)d_in[15];
    const float* b2      = (const float*)d_in[16];
    const float* Wf      = (const float*)d_in[17];
    const float* bf      = (const float*)d_in[18];

    float* outv = (float*)d_out;               // (B,NC)     @ 0
    float* hn   = outv + B_ * NC_;             // (B,H)      @ 160
    float* attn = hn + B_ * H_;                // (B,HEADS,T)@ 4256

    // Workspace layout (floats)
    float* ws      = (float*)d_ws;
    float* cat     = ws;                                   // (B*T, 2E)
    float* Mbuf    = cat     + (long)B_ * T_ * 2 * E_;     // (B,T,T)
    float* tmp1    = Mbuf    + (long)B_ * T_ * T_;         // (B*T, H)
    float* tmp2    = tmp1    + (long)B_ * T_ * H_;
    float* nextin  = tmp2    + (long)B_ * T_ * H_;
    float* xg      = nextin  + (long)B_ * T_ * H_;         // (B*T, 3H)
    float* outputs = xg      + (long)B_ * T_ * 3 * H_;     // (B*T, H)
    float* abuf    = outputs + (long)B_ * T_ * H_;         // (B*T, HEADS)
    float* Wc1t    = abuf    + (long)B_ * T_ * HEADS_;     // (2E, H)
    float* Wc2t    = Wc1t    + 2 * E_ * H_;
    float* Wiht    = Wc2t    + 2 * E_ * H_;                // (H, 3H)
    float* Whht    = Wiht    + H_ * 3 * H_;                // (H, 3H)
    float* W1t     = Whht    + H_ * 3 * H_;                // (H, H)

    // Weight transposes (once per call, tiny)
    transpose_kernel<<<(2 * E_ * H_ + 255) / 256, 256, 0, stream>>>(Wc1, Wc1t, H_, 2 * E_);
    transpose_kernel<<<(2 * E_ * H_ + 255) / 256, 256, 0, stream>>>(Wc2, Wc2t, H_, 2 * E_);
    transpose_kernel<<<(3 * H_ * H_ + 255) / 256, 256, 0, stream>>>(W_ih, Wiht, 3 * H_, H_);
    transpose_kernel<<<(3 * H_ * H_ + 255) / 256, 256, 0, stream>>>(W_hh, Whht, 3 * H_, H_);
    transpose_kernel<<<(H_ * H_ + 255) / 256, 256, 0, stream>>>(W1, W1t, H_, H_);

    // 1) Hemb -> cat[:, :E]
    embed_kernel<<<B_ * T_ / 4, 256, 0, stream>>>(x, emb, cat);

    // 2) Layer mix -> Mbuf (stored as M[b,j,i])
    mix_kernel<<<(B_ * T_ * T_ / 4 + 255) / 256, 256, 0, stream>>>(G, Gp, mix_w, Mbuf);

    // 3) weighted = M^T @ Hemb per batch  -> cat[:, E:]
    wmma_gemm_kernel<true><<<dim3(T_ / 64, E_ / 128, B_), 256, 0, stream>>>(
        Mbuf, cat, cat + E_, /*K=*/T_, /*lda=*/T_, /*ldb=*/2 * E_, /*ldc=*/2 * E_,
        (long)T_ * T_, (long)T_ * 2 * E_, (long)T_ * 2 * E_);

    // 4) SwiGLU projections: tmp1 = cat@Wc1^T, tmp2 = cat@Wc2^T
    wmma_gemm_kernel<false><<<dim3(B_ * T_ / 64, H_ / 128, 1), 256, 0, stream>>>(
        cat, Wc1t, tmp1, 2 * E_, 2 * E_, H_, H_, 0, 0, 0);
    wmma_gemm_kernel<false><<<dim3(B_ * T_ / 64, H_ / 128, 1), 256, 0, stream>>>(
        cat, Wc2t, tmp2, 2 * E_, 2 * E_, H_, H_, 0, 0, 0);
    swiglu_kernel<<<(B_ * T_ * H_ + 255) / 256, 256, 0, stream>>>(
        tmp1, tmp2, nextin, B_ * T_ * H_);

    // 5) GRU input projection: xg = next_in @ W_ih^T
    wmma_gemm_kernel<false><<<dim3(B_ * T_ / 64, 3 * H_ / 128, 1), 256, 0, stream>>>(
        nextin, Wiht, xg, H_, H_, 3 * H_, 3 * H_, 0, 0, 0);

    // 6) Sequential GRU scan (single persistent workgroup, WMMA per step)
    gru_scan_kernel<<<1, 768, 0, stream>>>(xg, Whht, b_ih, b_hh, h0, outputs, hn);

    // 7) Attention logits: tmp1 = tanh(outputs @ W1^T + b1), then heads
    wmma_gemm_kernel<false><<<dim3(B_ * T_ / 64, H_ / 128, 1), 256, 0, stream>>>(
        outputs, W1t, tmp1, H_, H_, H_, H_, 0, 0, 0);
    tanh_bias_kernel<<<(B_ * T_ * H_ + 255) / 256, 256, 0, stream>>>(tmp1, b1, B_ * T_ * H_);
    head_proj_kernel<<<(B_ * T_ * HEADS_ + 255) / 256, 256, 0, stream>>>(tmp1, W2, b2, abuf);

    // 8) Masked softmax -> attention (B,HEADS,T) directly into d_out
    softmax_kernel<<<B_ * HEADS_, 256, 0, stream>>>(abuf, lengths, attn);

    // 9) Pool + classifier -> output
    finalize_kernel<<<B_, 256, 0, stream>>>(attn, outputs, Wf, bf, outv);
}